// DynamicActionSpace_8555574854221
// MI455X (gfx1250) — compile-verified
//
#include <hip/hip_runtime.h>
#include <hip/hip_bf16.h>
#include <math.h>

// Problem constants (from reference)
#define Bsz   512
#define Hdim  2048
#define Aact  1024
#define Dmod  256
#define NHEAD 8
#define HDIM  32
#define MINK  64

typedef __attribute__((ext_vector_type(2))) float v2f;
typedef __attribute__((ext_vector_type(4))) float v4f;
typedef __attribute__((ext_vector_type(8))) float v8f;

// ---------------------------------------------------------------------------
// C[M,N] = A[M,K] @ W[N,K]^T + bias[N]   via V_WMMA_F32_16X16X4_F32
// One wave (32 threads) per 16x16 C tile; K stepped by 4.
// A frag (16x4 f32, 2 VGPR/lane): lanes 0-15 M=lane K={0,1}; lanes 16-31 K={2,3}
// B frag (4x16 f32): lanes 0-15 N=lane K={0,1}; lanes 16-31 K={2,3}
// C/D (8 VGPR): vgpr j -> M = j + 8*(lane>=16), N = lane&15
// ---------------------------------------------------------------------------
__global__ __launch_bounds__(32)
void gemm_nt_wmma(const float* __restrict__ A, const float* __restrict__ W,
                  const float* __restrict__ bias, float* __restrict__ C,
                  int M, int N, int K) {
  const int tn   = blockIdx.x;            // N/16 tiles
  const int tm   = blockIdx.y;            // M/16 tiles
  const int lane = threadIdx.x & 31;
  const int r    = lane & 15;
  const int kof  = (lane >> 4) * 2;       // 0 or 2

  const float* Arow = A + (size_t)(tm * 16 + r) * K;
  const float* Wrow = W + (size_t)(tn * 16 + r) * K;

  v8f acc = {};
  for (int k = 0; k < K; k += 4) {
    v2f a = { Arow[k + kof], Arow[k + kof + 1] };
    v2f b = { Wrow[k + kof], Wrow[k + kof + 1] };
    acc = __builtin_amdgcn_wmma_f32_16x16x4_f32(
        /*neg_a=*/false, a, /*neg_b=*/false, b,
        /*c_mod=*/(short)0, acc, /*reuse_a=*/false, /*reuse_b=*/false);
  }

  const int n     = tn * 16 + r;
  const int mbase = tm * 16 + (lane >> 4) * 8;
  const float bv  = bias ? bias[n] : 0.0f;
#pragma unroll
  for (int j = 0; j < 8; ++j)
    C[(size_t)(mbase + j) * N + n] = acc[j] + bv;
}

// ---------------------------------------------------------------------------
// per_action[a] = action_emb[a,:] . W_scorer[0,:] + b_scorer
// ---------------------------------------------------------------------------
__global__ void per_action_kernel(const float* __restrict__ emb,
                                  const float* __restrict__ Ws,
                                  const float* __restrict__ bs,
                                  float* __restrict__ out) {
  int a = blockIdx.x * blockDim.x + threadIdx.x;
  if (a >= Aact) return;
  const float* row = emb + (size_t)a * Dmod;
  float s = bs[0];
#pragma unroll 8
  for (int d = 0; d < Dmod; ++d) s += row[d] * Ws[d];
  out[a] = s;
}

// ---------------------------------------------------------------------------
// Per batch row: mean/std(ddof=1)/threshold -> k_per, full bitonic sort
// (descending score, ties by ascending index == jax.lax.top_k order),
// emit gather indices (invalid -> 0) and softmax(pruned_scores) weights.
// One block of 512 threads per row; 1024 elements in LDS.
// ---------------------------------------------------------------------------
__global__ __launch_bounds__(512)
void rank_kernel(const float* __restrict__ scores,
                 float* __restrict__ weights_out,
                 int* __restrict__ idx_out) {
  const int b = blockIdx.x;
  const int t = threadIdx.x;
  __shared__ float sS[Aact];
  __shared__ int   sI[Aact];
  __shared__ float red[512];

  const float* row = scores + (size_t)b * Aact;
  float x0 = row[t], x1 = row[t + 512];
  sS[t] = x0; sS[t + 512] = x1;
  sI[t] = t;  sI[t + 512] = t + 512;

  // mean
  red[t] = x0 + x1;
  __syncthreads();
  for (int s = 256; s > 0; s >>= 1) { if (t < s) red[t] += red[t + s]; __syncthreads(); }
  const float mean = red[0] * (1.0f / (float)Aact);
  __syncthreads();
  // var (ddof=1)
  float d0 = x0 - mean, d1 = x1 - mean;
  red[t] = d0 * d0 + d1 * d1;
  __syncthreads();
  for (int s = 256; s > 0; s >>= 1) { if (t < s) red[t] += red[t + s]; __syncthreads(); }
  const float thr = mean + 0.1f * sqrtf(red[0] / (float)(Aact - 1));
  __syncthreads();
  // count above threshold
  red[t] = (x0 > thr ? 1.0f : 0.0f) + (x1 > thr ? 1.0f : 0.0f);
  __syncthreads();
  for (int s = 256; s > 0; s >>= 1) { if (t < s) red[t] += red[t + s]; __syncthreads(); }
  int kper = (int)red[0];
  if (kper < MINK) kper = MINK;
  __syncthreads();

  // bitonic sort: final order = descending score, ties -> ascending index
  for (unsigned ksz = 2; ksz <= Aact; ksz <<= 1) {
    for (unsigned j = ksz >> 1; j > 0; j >>= 1) {
      __syncthreads();
#pragma unroll
      for (int rep = 0; rep < 2; ++rep) {
        unsigned i   = (unsigned)t + (unsigned)rep * 512u;
        unsigned ixj = i ^ j;
        if (ixj > i) {
          float si = sS[i], sj = sS[ixj];
          int   ii = sI[i], ij = sI[ixj];
          bool i_first   = (si > sj) || (si == sj && ii < ij);
          bool ascending = ((i & ksz) == 0);
          if (ascending ? !i_first : i_first) {
            sS[i] = sj; sS[ixj] = si;
            sI[i] = ij; sI[ixj] = ii;
          }
        }
      }
    }
  }
  __syncthreads();

  // prune + indices
  float p0 = (t       < kper) ? sS[t]       : 0.0f;
  float p1 = (t + 512 < kper) ? sS[t + 512] : 0.0f;
  int   i0 = (t       < kper) ? sI[t]       : 0;
  int   i1 = (t + 512 < kper) ? sI[t + 512] : 0;
  idx_out[(size_t)b * Aact + t]       = i0;
  idx_out[(size_t)b * Aact + t + 512] = i1;

  // softmax over pruned scores (zeros participate)
  red[t] = fmaxf(p0, p1);
  __syncthreads();
  for (int s = 256; s > 0; s >>= 1) { if (t < s) red[t] = fmaxf(red[t], red[t + s]); __syncthreads(); }
  const float mx = red[0];
  __syncthreads();
  float e0 = __expf(p0 - mx), e1 = __expf(p1 - mx);
  red[t] = e0 + e1;
  __syncthreads();
  for (int s = 256; s > 0; s >>= 1) { if (t < s) red[t] += red[t + s]; __syncthreads(); }
  const float inv = 1.0f / red[0];
  weights_out[(size_t)b * Aact + t]       = e0 * inv;
  weights_out[(size_t)b * Aact + t + 512] = e1 * inv;
}

// ---------------------------------------------------------------------------
// Attention per batch: 8 waves = 8 heads. Keys/values gathered from
// K_all/V_all (L2-resident) via LDS-staged indices. Wave32 shuffle
// reductions; unnormalized exps cached in LDS (1/sum applied at the end) so the V pass is
// one LDS broadcast + one coalesced 128B read + one FMA per key.
// ---------------------------------------------------------------------------
__global__ __launch_bounds__(256)
void attn_kernel(const float* __restrict__ q, const float* __restrict__ Kall,
                 const float* __restrict__ Vall, const int* __restrict__ idx,
                 float* __restrict__ ctx_att) {
  const int b    = blockIdx.x;
  const int head = threadIdx.x >> 5;
  const int lane = threadIdx.x & 31;
  __shared__ float qs[Dmod];
  __shared__ int   sidx[Aact];
  __shared__ float lg[NHEAD][Aact];

  qs[threadIdx.x] = q[(size_t)b * Dmod + threadIdx.x];
#pragma unroll
  for (int c = 0; c < Aact / 256; ++c)
    sidx[c * 256 + threadIdx.x] = idx[(size_t)b * Aact + c * 256 + threadIdx.x];
  __syncthreads();

  const float* qh    = qs + head * HDIM;
  const float  scale = 0.17677669529663687f; // 1/sqrt(32)

  float mx = -3.0e38f;
  for (int c = 0; c < Aact / 32; ++c) {
    int a = c * 32 + lane;
    const float* krow = Kall + (size_t)sidx[a] * Dmod + head * HDIM;
    float dot = 0.0f;
#pragma unroll
    for (int d = 0; d < HDIM; ++d) dot += qh[d] * krow[d];
    dot *= scale;
    lg[head][a] = dot;
    mx = fmaxf(mx, dot);
  }
  for (int m = 16; m > 0; m >>= 1) mx = fmaxf(mx, __shfl_xor(mx, m, 32));

  float sum = 0.0f;
  for (int c = 0; c < Aact / 32; ++c) {
    int a = c * 32 + lane;
    float e = __expf(lg[head][a] - mx);
    lg[head][a] = e;                     // cache exp in LDS
    sum += e;
  }
  for (int m = 16; m > 0; m >>= 1) sum += __shfl_xor(sum, m, 32);
  const float inv = 1.0f / sum;

  // lane == head dim; V rows are coalesced 128B reads from L2
  float acc = 0.0f;
  for (int a = 0; a < Aact; ++a) {
    float w = lg[head][a];               // LDS broadcast (same addr all lanes)
    acc += w * Vall[(size_t)sidx[a] * Dmod + head * HDIM + lane];
  }
  ctx_att[(size_t)b * Dmod + head * HDIM + lane] = acc * inv;
}

// ---------------------------------------------------------------------------
// out[b,a,:] = attended[b,:] + action_emb[idx[b,a],:]
// 512 MB write-once stream -> nontemporal b128 stores (TH=NT) so we don't
// write-allocate the output through WGP$/L2; sources stay RT (L2-resident).
// Uses clang ext_vector float4 (v4f) — __builtin_nontemporal_store rejects
// HIP's float4 class type.
// ---------------------------------------------------------------------------
__global__ __launch_bounds__(64)
void out_add_kernel(const float* __restrict__ emb, const float* __restrict__ attended,
                    const int* __restrict__ idx, float* __restrict__ out) {
  const int a = blockIdx.x;
  const int b = blockIdx.y;
  const int t = threadIdx.x;                      // 64 threads * 16B = 256 floats
  const int r = idx[(size_t)b * Aact + a];
  const v4f* src = (const v4f*)(emb + (size_t)r * Dmod);
  const v4f* att = (const v4f*)(attended + (size_t)b * Dmod);
  v4f* dst = (v4f*)(out + ((size_t)b * Aact + a) * Dmod);
  v4f val = src[t] + att[t];
  __builtin_nontemporal_store(val, dst + t);
}

// ---------------------------------------------------------------------------
extern "C" void kernel_launch(void* const* d_in, const int* in_sizes, int n_in,
                              void* d_out, int out_size, void* d_ws, size_t ws_size,
                              hipStream_t stream) {
  const float* context    = (const float*)d_in[0];
  const float* action_emb = (const float*)d_in[1];
  const float* W_ctx      = (const float*)d_in[2];
  const float* b_ctx      = (const float*)d_in[3];
  const float* W_scorer   = (const float*)d_in[4];
  const float* b_scorer   = (const float*)d_in[5];
  const float* in_proj_w  = (const float*)d_in[6];
  const float* in_proj_b  = (const float*)d_in[7];
  const float* out_proj_w = (const float*)d_in[8];
  const float* out_proj_b = (const float*)d_in[9];
  float* out = (float*)d_out;
  float* weights_out = out + (size_t)Bsz * Aact * Dmod;  // second tuple output

  // workspace carve-up (floats)
  float* ws       = (float*)d_ws;
  float* ctx_proj = ws;                                  // 512*256
  float* scores   = ctx_proj + Bsz * Dmod;               // 512*1024
  float* per_act  = scores   + Bsz * Aact;               // 1024
  float* qbuf     = per_act  + Aact;                     // 512*256
  float* Kall     = qbuf     + Bsz * Dmod;               // 1024*256
  float* Vall     = Kall     + Aact * Dmod;              // 1024*256
  float* ctx_att  = Vall     + Aact * Dmod;              // 512*256
  float* attended = ctx_att  + Bsz * Dmod;               // 512*256
  int*   topidx   = (int*)(attended + Bsz * Dmod);       // 512*1024 ints

  // 1) ctx_proj = context @ W_ctx^T + b_ctx          (512 x 256, K=2048)
  gemm_nt_wmma<<<dim3(Dmod / 16, Bsz / 16), 32, 0, stream>>>(
      context, W_ctx, b_ctx, ctx_proj, Bsz, Dmod, Hdim);

  // 2) per_action scorer bias
  per_action_kernel<<<Aact / 256, 256, 0, stream>>>(action_emb, W_scorer, b_scorer, per_act);

  // 3) scores = ctx_proj @ action_emb^T + per_action (512 x 1024, K=256)
  gemm_nt_wmma<<<dim3(Aact / 16, Bsz / 16), 32, 0, stream>>>(
      ctx_proj, action_emb, per_act, scores, Bsz, Aact, Dmod);

  // 4) stats + top_k sort + pruned-softmax weights + gather indices
  rank_kernel<<<Bsz, 512, 0, stream>>>(scores, weights_out, topidx);

  // 5) q / K_all / V_all projections (gather-free: project the 1024-row table)
  gemm_nt_wmma<<<dim3(Dmod / 16, Bsz / 16), 32, 0, stream>>>(
      ctx_proj, in_proj_w, in_proj_b, qbuf, Bsz, Dmod, Dmod);
  gemm_nt_wmma<<<dim3(Dmod / 16, Aact / 16), 32, 0, stream>>>(
      action_emb, in_proj_w + (size_t)Dmod * Dmod, in_proj_b + Dmod, Kall, Aact, Dmod, Dmod);
  gemm_nt_wmma<<<dim3(Dmod / 16, Aact / 16), 32, 0, stream>>>(
      action_emb, in_proj_w + (size_t)2 * Dmod * Dmod, in_proj_b + 2 * Dmod, Vall, Aact, Dmod, Dmod);

  // 6) gathered attention (indices into K_all/V_all, L2-resident)
  attn_kernel<<<Bsz, 256, 0, stream>>>(qbuf, Kall, Vall, topidx, ctx_att);

  // 7) attended = ctx_att @ out_proj_w^T + out_proj_b
  gemm_nt_wmma<<<dim3(Dmod / 16, Bsz / 16), 32, 0, stream>>>(
      ctx_att, out_proj_w, out_proj_b, attended, Bsz, Dmod, Dmod);

  // 8) 512 MB streaming output: attended[b] + action_emb[idx[b,a]]
  out_add_kernel<<<dim3(Aact, Bsz), 64, 0, stream>>>(action_emb, attended, topidx, out);
}